// xDeepFM_43224550867494
// MI455X (gfx1250) — compile-verified
//
#include <hip/hip_runtime.h>
#include <math.h>

// xDeepFM CIN on gfx1250 (wave32, WMMA, TDM).
// Sizes: BS=32, NUM_TREES=NF=256, EMB=64, 3 layers.
#define BS     32
#define TS     256              // num_trees == num_filters
#define EMBD   64
#define KSPLIT 2                // split step1 contraction across blocks (latency hiding)
#define KRANGE (TS / KSPLIT)    // 128 y-values per block
#define LDSB   (KRANGE + 4)     // padded LDS row stride (floats): bank-conflict-free

typedef float v2f __attribute__((ext_vector_type(2)));
typedef float v8f __attribute__((ext_vector_type(8)));
typedef unsigned int u32x4 __attribute__((ext_vector_type(4)));
typedef int i32x4 __attribute__((ext_vector_type(4)));
typedef int i32x8 __attribute__((ext_vector_type(8)));

// Tensor Data Mover builtin: 5-arg form on ROCm 7.2 (clang-22).
#if defined(__clang_major__) && (__clang_major__ < 23)
#  if __has_builtin(__builtin_amdgcn_tensor_load_to_lds)
#    define USE_TDM 1
#  endif
#endif
#ifndef USE_TDM
#  define USE_TDM 0
#endif

static __device__ __forceinline__ v8f wmma4(v2f a, v2f b, v8f c) {
  // D(16x16) = A(16x4) * B(4x16) + C, fp32 WMMA (VOP3P), wave32.
  return __builtin_amdgcn_wmma_f32_16x16x4_f32(false, a, false, b, (short)0, c,
                                               false, false);
}

// ---------------------------------------------------------------------------
// K1: gather embeddings. Writes
//   X0T[f][b][t] : transposed x0 (step1 stages contiguous rows)
//   X0P[b][t'][p]: the "reshape view" operand x0p[b,t',p] = x0[b, 64*(t'%4)+p, t'/4]
__global__ void __launch_bounds__(256)
k_gather(const int* __restrict__ x, const float* __restrict__ emb,
         float* __restrict__ X0T, float* __restrict__ X0P)
{
  const int b = blockIdx.x;
  const int t = threadIdx.x;
  const int idx = x[b * TS + t];
  const float* row = emb + (size_t)idx * EMBD;
  const int q = t >> 6, pr = t & 63;
  for (int f = 0; f < EMBD; f += 4) {
    const float4 rv = *(const float4*)(row + f);
    const float vv[4] = {rv.x, rv.y, rv.z, rv.w};
#pragma unroll
    for (int j = 0; j < 4; ++j) {
      X0T[((size_t)(f + j) * BS + b) * TS + t] = vv[j];                 // coalesced in t
      X0P[((size_t)b * TS + (4 * (f + j) + q)) * EMBD + pr] = vv[j];    // coalesced in p
    }
  }
}

// ---------------------------------------------------------------------------
// K2 (step 1, per layer): G[kh][t'][b][h] = sum_{y in kh-half} XT[t'/4][b][y]*W[h][t'][y]
// Grid (256, KSPLIT): one block per (t', k-half) -> 512 blocks in flight for the
// HBM-bound W stream. xk slab staged via the Tensor Data Mover (padded LDS tile).
__global__ void __launch_bounds__(256)
k_step1(const float* __restrict__ XT,   // [64][32][256]
        const float* __restrict__ W,    // [256][256][256] = [h][t'][y]
        float* __restrict__ G)          // [KSPLIT][256][32][256]
{
  __shared__ __align__(16) float Xls[BS * LDSB];
  const int tp  = blockIdx.x;           // t'
  const int kh  = blockIdx.y;           // k-half
  const int f   = tp >> 2;
  const int tid = threadIdx.x;
  const int y0  = kh * KRANGE;

#if USE_TDM
  if (tid < 32) {  // TDM ignores EXEC; issue exactly one DMA from wave 0
    const unsigned long long ga =
        (unsigned long long)(uintptr_t)XT +
        ((unsigned long long)f * (BS * TS) + (unsigned)y0) * 4ull;
    // D# group0: count=1 | lds_addr | global_addr | type=2 ("image")
    const u32x4 g0 = {1u,
                      (unsigned)(uintptr_t)(void*)Xls,
                      (unsigned)(ga & 0xFFFFFFFFull),
                      (unsigned)((ga >> 32) & 0x1FFFFFFull) | (2u << 30)};
    // D# group1: data_size=4B, pad_enable, pad_interval=128 DW, pad_amount=4 DW;
    // tensor_dim0=0x40000000 (no OOB), tensor_dim1=32, tile=128x32, dim0_stride=256.
    const i32x8 g1 = {(2 << 16) | (1 << 20) | (6 << 22) | (3 << 25),
                      0,
                      0x4000 | (32 << 16),
                      (KRANGE << 16),
                      32,
                      TS,
                      0,
                      0};
    const i32x4 z4 = {0, 0, 0, 0};
    __builtin_amdgcn_tensor_load_to_lds(g0, g1, z4, z4, 0);
    __builtin_amdgcn_s_wait_tensorcnt(0);
  }
#else
  { // manual staging fallback (same padded layout)
    const float* src = XT + (size_t)f * (BS * TS) + y0;
#pragma unroll
    for (int i = 0; i < 4; ++i) {
      const int e = (tid + i * 256) * 4;        // 0..4092
      const int bb = e >> 7, y = e & 127;
      *(float4*)(Xls + bb * LDSB + y) = *(const float4*)(src + bb * TS + y);
    }
  }
#endif
  __syncthreads();

  const int wave = tid >> 5, lane = tid & 31;
  const int lm = lane & 15;             // M / N position within tile
  const int kb = (lane >> 4) * 2;       // half-wave k offset (K=0,1 | K=2,3)
  const int h0 = wave * 32;             // this wave owns h in [h0, h0+32)

  const float* w0 = W + ((size_t)(h0 + lm) * TS + tp) * TS + y0;       // h-tile 0
  const float* w1 = W + ((size_t)(h0 + 16 + lm) * TS + tp) * TS + y0;  // h-tile 1

  v8f c00 = {0,0,0,0,0,0,0,0}, c01 = c00, c10 = c00, c11 = c00;

#pragma unroll 8
  for (int k = 0; k < KRANGE; k += 4) {
    const int kk = k + kb;
    if ((k & 31) == 0) {                               // one prefetch per 8 steps
      __builtin_prefetch(w0 + ((kk + 64) & (KRANGE - 1)), 0, 1);
      __builtin_prefetch(w1 + ((kk + 64) & (KRANGE - 1)), 0, 1);
    }
    // A frags (b-rows) from LDS: lane L -> row lm (+16), K pair at kk
    const v2f a0 = *(const v2f*)(Xls + lm * LDSB + kk);
    const v2f a1 = *(const v2f*)(Xls + (16 + lm) * LDSB + kk);
    // B frags (h-cols) from W, read-once stream -> non-temporal
    const v2f b0 = __builtin_nontemporal_load((const v2f*)(w0 + kk));
    const v2f b1 = __builtin_nontemporal_load((const v2f*)(w1 + kk));
    c00 = wmma4(a0, b0, c00);
    c01 = wmma4(a0, b1, c01);
    c10 = wmma4(a1, b0, c10);
    c11 = wmma4(a1, b1, c11);
  }

  // C/D layout: VGPR r -> rows r (lanes 0-15) and r+8 (lanes 16-31), col = lm
  float* g = G + (size_t)(kh * TS + tp) * (BS * TS);
  const int n0 = h0 + lm, n1 = n0 + 16;
  const int hi8 = (lane >> 4) * 8;
#pragma unroll
  for (int r = 0; r < 8; ++r) {
    const int m = hi8 + r;                       // b row
    g[(size_t)m * TS + n0]        = c00[r];
    g[(size_t)m * TS + n1]        = c01[r];
    g[(size_t)(m + 16) * TS + n0] = c10[r];
    g[(size_t)(m + 16) * TS + n1] = c11[r];
  }
}

// ---------------------------------------------------------------------------
// K3 (step 2, per layer): per batch b,
//   xk_next[h][p] = sum_{j<512} G[j][b][h] * X0P[b][j&255][p] + bias[h]
// (partial-G halves merged for free through the 512-deep WMMA accumulation).
// Writes XKT[p][b][h] (transposed -> next layer's step1 tile is row-contiguous)
// and outs[b][p] = sum_h xk_next[h][p] via a deterministic LDS reduction.
__global__ void __launch_bounds__(256)
k_step2(const float* __restrict__ G,     // [KSPLIT*256][b][h]
        const float* __restrict__ X0P,   // [b][t'][p]
        const float* __restrict__ bias,  // [256]
        float* __restrict__ XKT,         // [p][b][h]
        float* __restrict__ outs)        // [b][64]
{
  __shared__ float bias_s[TS];
  __shared__ float red[8 * 4 * 32];
  const int b = blockIdx.x;
  const int tid = threadIdx.x;
  bias_s[tid] = bias[tid];
  __syncthreads();

  const int wave = tid >> 5, lane = tid & 31;
  const int lm = lane & 15, kb = (lane >> 4) * 2;
  const int m0 = wave * 32;                    // h-tile base (2 tiles of 16)

  v8f acc[2][4] = {};
  const float* gb = G + (size_t)b * TS;        // + j*8192 + h
  const float* xp = X0P + (size_t)b * (TS * EMBD);

#pragma unroll 4
  for (int k = 0; k < KSPLIT * TS; k += 4) {
    const int ka = k + kb;
    const int ta = ka & (TS - 1);              // t' index for X0P
    v2f a0, a1;                                // A = G_b^T : M=h, K=j
    a0.x = gb[(size_t)ka * (BS * TS) + m0 + lm];
    a0.y = gb[(size_t)(ka + 1) * (BS * TS) + m0 + lm];
    a1.x = gb[(size_t)ka * (BS * TS) + m0 + 16 + lm];
    a1.y = gb[(size_t)(ka + 1) * (BS * TS) + m0 + 16 + lm];
#pragma unroll
    for (int nt = 0; nt < 4; ++nt) {           // p in 4 tiles of 16
      v2f bf;
      bf.x = xp[(size_t)ta * EMBD + nt * 16 + lm];
      bf.y = xp[(size_t)(ta + 1) * EMBD + nt * 16 + lm];
      acc[0][nt] = wmma4(a0, bf, acc[0][nt]);
      acc[1][nt] = wmma4(a1, bf, acc[1][nt]);
    }
  }

  const int hi8 = (lane >> 4) * 8;
  float ssum[4] = {0.f, 0.f, 0.f, 0.f};
#pragma unroll
  for (int mt = 0; mt < 2; ++mt) {
#pragma unroll
    for (int nt = 0; nt < 4; ++nt) {
      const int p = nt * 16 + lm;
      float* dst = XKT + ((size_t)p * BS + b) * TS;
#pragma unroll
      for (int r = 0; r < 8; ++r) {
        const int h = m0 + mt * 16 + hi8 + r;
        const float v = acc[mt][nt][r] + bias_s[h];
        dst[h] = v;
        ssum[nt] += v;                          // partial sum over 16 h's
      }
    }
  }
#pragma unroll
  for (int nt = 0; nt < 4; ++nt) red[(wave * 4 + nt) * 32 + lane] = ssum[nt];
  __syncthreads();
  if (tid < EMBD) {                             // p = tid; deterministic reduce
    const int nt = tid >> 4, li = tid & 15;
    float tot = 0.f;
#pragma unroll
    for (int w = 0; w < 8; ++w)
      tot += red[(w * 4 + nt) * 32 + li] + red[(w * 4 + nt) * 32 + li + 16];
    outs[b * EMBD + tid] = tot;
  }
}

// ---------------------------------------------------------------------------
// K4: dot([outs1|outs2|outs3|x0.flat], ffn_w) + ffn_b -> sigmoid. One block per b.
__global__ void __launch_bounds__(256)
k_final(const float* __restrict__ X0T, const float* __restrict__ OUTS,
        const float* __restrict__ fw, const float* __restrict__ fb,
        float* __restrict__ out)
{
  __shared__ float red[256];
  const int b = blockIdx.x, tid = threadIdx.x;
  float s = 0.f;
  if (tid < 3 * EMBD)   // layer outputs: index = layer*64 + p
    s += OUTS[(tid >> 6) * (BS * EMBD) + b * EMBD + (tid & 63)] * fw[tid];
  // wide part: flat index j = t*64 + f, t = tid (X0T reads coalesced in t)
#pragma unroll 4
  for (int f = 0; f < EMBD; ++f) {
    const float xv = X0T[((size_t)f * BS + b) * TS + tid];
    s += xv * fw[3 * EMBD + tid * EMBD + f];
  }
  red[tid] = s;
  __syncthreads();
  for (int off = 128; off > 0; off >>= 1) {
    if (tid < off) red[tid] += red[tid + off];
    __syncthreads();
  }
  if (tid == 0) {
    const float d = red[0] + fb[0];
    out[b] = 1.f / (1.f + __expf(-d));
  }
}

// ---------------------------------------------------------------------------
extern "C" void kernel_launch(void* const* d_in, const int* in_sizes, int n_in,
                              void* d_out, int out_size, void* d_ws, size_t ws_size,
                              hipStream_t stream) {
  (void)in_sizes; (void)n_in; (void)out_size; (void)ws_size;
  const int*   x   = (const int*)d_in[0];
  const float* emb = (const float*)d_in[1];
  const float* Wl[3] = {(const float*)d_in[2], (const float*)d_in[4],
                        (const float*)d_in[6]};
  const float* bl[3] = {(const float*)d_in[3], (const float*)d_in[5],
                        (const float*)d_in[7]};
  const float* fw  = (const float*)d_in[8];
  const float* fb  = (const float*)d_in[9];
  float* out = (float*)d_out;

  // Workspace layout (~22 MB; G is L2-resident so the KSPLIT partials never
  // cost HBM bandwidth):
  char* ws = (char*)d_ws;
  float* X0T  = (float*)(ws);                          //  2 MB [f][b][t]
  float* X0P  = (float*)(ws + ((size_t)2 << 20));      //  2 MB [b][t'][p]
  float* XKT  = (float*)(ws + ((size_t)4 << 20));      //  2 MB [p][b][h]
  float* G    = (float*)(ws + ((size_t)6 << 20));      // 16 MB [kh][t'][b][h]
  float* OUTS = (float*)(ws + ((size_t)22 << 20));     // 24 KB [layer][b][p]

  k_gather<<<BS, 256, 0, stream>>>(x, emb, X0T, X0P);
  const float* xt = X0T;
  for (int l = 0; l < 3; ++l) {
    k_step1<<<dim3(TS, KSPLIT), 256, 0, stream>>>(xt, Wl[l], G);
    k_step2<<<BS, 256, 0, stream>>>(G, X0P, bl[l], XKT,
                                    OUTS + (size_t)l * BS * EMBD);
    xt = XKT;
  }
  k_final<<<BS, 256, 0, stream>>>(X0T, OUTS, fw, fb, out);
}